// GraphMessageLayer_28355374088952
// MI455X (gfx1250) — compile-verified
//
#include <hip/hip_runtime.h>
#include <hip/hip_bf16.h>
#include <math.h>

// ---------------------------------------------------------------------------
// E(n)-GNN message layer, algebraically factored:
//   hidden(b,i,j) = A[b,i] + C[b,j] + dist_ij * w_d + be1   (A,C are WMMA GEMMs)
//   S[b,i] = sum_j mask * silu(hidden)                      (VALU pass, LDS dist)
//   agg    = (S @ We2 + denom*be2) / max(denom,1)           (WMMA GEMM, epilogue)
//   out    = ns + (silu(ns@Wn1t + agg@Wn1b + bn1) @ Wn2 + bn2) * mask  (WMMA)
// All matrix math on V_WMMA_F32_16X16X4_F32 (fp32, matches reference dtype).
// ---------------------------------------------------------------------------

typedef __attribute__((ext_vector_type(2))) float v2f;
typedef __attribute__((ext_vector_type(8))) float v8f;

constexpr int Bc = 8, Nc = 256, Hc = 128, BN = Bc * Nc;

// One wave32 computes one 16x16 fp32 output tile; 4 waves/block along N.
// A: [M x 128] row-major, W: [128 x 128] row-major ([in,out] as stored).
// MODE 0: out = acc + bias(optional)
// MODE 1: out = (acc + denom[m]*bias[n]) / max(denom[m],1)
// MODE 2: dual-K accumulate (A2@W2), out = silu(acc + bias)
// MODE 3: out = resid[m,n] + (acc + bias[n]) * maskp[m]
template <int MODE>
__global__ __launch_bounds__(128) void gemm_wmma_f32(
    const float* __restrict__ A1, const float* __restrict__ W1,
    const float* __restrict__ A2, const float* __restrict__ W2,
    const float* __restrict__ bias, const float* __restrict__ denom,
    const float* __restrict__ resid, const float* __restrict__ maskp,
    float* __restrict__ Out)
{
    const int wave = threadIdx.x >> 5;
    const int lane = threadIdx.x & 31;
    const int n0   = (blockIdx.x * 4 + wave) * 16;
    const int m0   = blockIdx.y * 16;
    const int mrow = lane & 15;          // M within tile (A operand)
    const int ncol = lane & 15;          // N within tile (B/C/D operands)
    const int kh   = (lane >> 4) << 1;   // lanes 0-15: K=k+{0,1}; 16-31: K=k+{2,3}

    v8f acc = {0.f, 0.f, 0.f, 0.f, 0.f, 0.f, 0.f, 0.f};

    {
        const float* ap = A1 + (m0 + mrow) * Hc + kh;
        const float* wp = W1 + kh * Hc + n0 + ncol;
        for (int k = 0; k < Hc; k += 4) {
            v2f av, bv;
            av.x = ap[k];
            av.y = ap[k + 1];
            bv.x = wp[k * Hc];
            bv.y = wp[(k + 1) * Hc];
            acc = __builtin_amdgcn_wmma_f32_16x16x4_f32(
                false, av, false, bv, (short)0, acc, false, false);
        }
    }
    if constexpr (MODE == 2) {           // second K block: agg @ Wn1[128:256]
        const float* ap = A2 + (m0 + mrow) * Hc + kh;
        const float* wp = W2 + kh * Hc + n0 + ncol;
        for (int k = 0; k < Hc; k += 4) {
            v2f av, bv;
            av.x = ap[k];
            av.y = ap[k + 1];
            bv.x = wp[k * Hc];
            bv.y = wp[(k + 1) * Hc];
            acc = __builtin_amdgcn_wmma_f32_16x16x4_f32(
                false, av, false, bv, (short)0, acc, false, false);
        }
    }

    const int n  = n0 + ncol;
    const int mb = m0 + ((lane >> 4) << 3);   // C/D: VGPR r -> M = r (+8 hi half)
    const float bn = bias ? bias[n] : 0.f;
#pragma unroll
    for (int r = 0; r < 8; ++r) {
        const int m = mb + r;
        float v = acc[r];
        if constexpr (MODE == 0) {
            v += bn;
        } else if constexpr (MODE == 1) {
            const float dr = denom[m];
            v = (v + dr * bn) / fmaxf(dr, 1.f);
        } else if constexpr (MODE == 2) {
            v += bn;
            v = v / (1.f + __expf(-v));          // SiLU
        } else {
            v += bn;
            v = resid[m * Hc + n] + v * maskp[m];
        }
        Out[m * Hc + n] = v;
    }
}

// One block per (b,i): S[b,i,c] = mask_i * sum_j mask_j * silu(A_i + C_j + d_ij*w_d)
// Distances computed once per block into LDS (2 sqrt/thread, then broadcast reads).
__global__ __launch_bounds__(128) void edge_reduce(
    const float* __restrict__ Amat, const float* __restrict__ Cmat,
    const float* __restrict__ pos, const float* __restrict__ maskp,
    const float* __restrict__ wd,
    float* __restrict__ S, float* __restrict__ denomraw)
{
    const int bi = blockIdx.x;          // b*N + i
    const int b  = bi >> 8;             // N = 256
    const int c  = threadIdx.x;         // channel 0..127

    __shared__ float dist_s[Nc];
    __shared__ float msk_s[Nc];

    const float pix = pos[bi * 3 + 0];
    const float piy = pos[bi * 3 + 1];
    const float piz = pos[bi * 3 + 2];
    for (int j = c; j < Nc; j += 128) {
        const int gj = (b << 8) + j;
        const float dx = pix - pos[gj * 3 + 0];
        const float dy = piy - pos[gj * 3 + 1];
        const float dz = piz - pos[gj * 3 + 2];
        const float sq = dx * dx + dy * dy + dz * dz;
        dist_s[j] = sq > 0.f ? sqrtf(sq) : 0.f;   // safe-sqrt, exact fwd match
        msk_s[j]  = maskp[gj];
    }
    __syncthreads();

    const float a = Amat[bi * Hc + c];            // includes be1
    const float w = wd[c];                        // We1 row 256 (distance column)
    const float* crow = Cmat + ((b << 8) * Hc + c);

    float accv = 0.f, msum = 0.f;
    for (int j = 0; j < Nc; ++j) {
        const float h = a + crow[j * Hc] + dist_s[j] * w;
        const float s = h / (1.f + __expf(-h));   // SiLU
        accv += msk_s[j] * s;
        msum += msk_s[j];
    }
    const float mi = maskp[bi];
    S[bi * Hc + c] = mi * accv;
    if (c == 0) denomraw[bi] = mi * msum;
}

extern "C" void kernel_launch(void* const* d_in, const int* in_sizes, int n_in,
                              void* d_out, int out_size, void* d_ws, size_t ws_size,
                              hipStream_t stream)
{
    const float* ns   = (const float*)d_in[0];   // [B,N,H]
    const float* pos  = (const float*)d_in[1];   // [B,N,3]
    const float* mask = (const float*)d_in[2];   // [B,N]
    const float* We1  = (const float*)d_in[3];   // [257,128]
    const float* be1  = (const float*)d_in[4];   // [128]
    const float* We2  = (const float*)d_in[5];   // [128,128]
    const float* be2  = (const float*)d_in[6];   // [128]
    const float* Wn1  = (const float*)d_in[7];   // [256,128]
    const float* bn1  = (const float*)d_in[8];   // [128]
    const float* Wn2  = (const float*)d_in[9];   // [128,128]
    const float* bn2  = (const float*)d_in[10];  // [128]
    float* out = (float*)d_out;

    float* ws  = (float*)d_ws;
    float* A   = ws;                // [BN,H]  ns@We1[0:H] + be1
    float* Cm  = A + BN * Hc;       // [BN,H]  ns@We1[H:2H]
    float* S   = Cm + BN * Hc;      // [BN,H]  masked silu-sum
    float* dnm = S + BN * Hc;       // [BN]    raw denom
    float* agg = dnm + BN;          // [BN,H]
    float* Hn  = agg + BN * Hc;     // [BN,H]
    (void)in_sizes; (void)n_in; (void)out_size; (void)ws_size;

    const dim3 gg(2, BN / 16), bb(128);

    // A = ns @ We1[0:128] + be1
    gemm_wmma_f32<0><<<gg, bb, 0, stream>>>(ns, We1, nullptr, nullptr,
                                            be1, nullptr, nullptr, nullptr, A);
    // Cm = ns @ We1[128:256]
    gemm_wmma_f32<0><<<gg, bb, 0, stream>>>(ns, We1 + Hc * Hc, nullptr, nullptr,
                                            nullptr, nullptr, nullptr, nullptr, Cm);
    // S, denom over all edges (w_d = We1 row 256)
    edge_reduce<<<dim3(BN), bb, 0, stream>>>(A, Cm, pos, mask,
                                             We1 + 2 * Hc * Hc, S, dnm);
    // agg = (S @ We2 + denom*be2) / max(denom,1)
    gemm_wmma_f32<1><<<gg, bb, 0, stream>>>(S, We2, nullptr, nullptr,
                                            be2, dnm, nullptr, nullptr, agg);
    // Hn = silu(ns @ Wn1[0:128] + agg @ Wn1[128:256] + bn1)
    gemm_wmma_f32<2><<<gg, bb, 0, stream>>>(ns, Wn1, agg, Wn1 + Hc * Hc,
                                            bn1, nullptr, nullptr, nullptr, Hn);
    // out = ns + (Hn @ Wn2 + bn2) * mask
    gemm_wmma_f32<3><<<gg, bb, 0, stream>>>(Hn, Wn2, nullptr, nullptr,
                                            bn2, nullptr, ns, mask, out);
}